// G_Net_70514773066441
// MI455X (gfx1250) — compile-verified
//
#include <hip/hip_runtime.h>
#include <hip/hip_bf16.h>

// MoE gathered-expert linear: out[b,:] = W[label[b]] @ x[b] + bias[label[b]]
// B=32768, IN=5, OUT=784, E=10.  Store-bandwidth bound (103 MB out @ 23.3 TB/s
// ~= 4.4 us floor). Compute via V_WMMA_F32_16X16X4_F32 on 16x16 output tiles,
// looping all 10 experts per tile and selecting rows by label (dense stores).
// All conditional values are selected AFTER unconditional loads so EXEC stays
// all-1s through the whole kernel (no saveexec around the WMMA loop).

typedef __attribute__((ext_vector_type(2))) float v2f;
typedef __attribute__((ext_vector_type(8))) float v8f;

#define N_IDEAS 5
#define N_EXPERTS 10
#define OUT_DIM 784
#define BATCH 32768

__global__ __launch_bounds__(256) void moe_wmma_kernel(
    const float* __restrict__ x,       // [B, 5]
    const int*   __restrict__ labels,  // [B]
    const float* __restrict__ W,       // [E, 784, 5]
    const float* __restrict__ bias,    // [E, 784]
    float*       __restrict__ out)     // [B, 784]
{
    const int wave = threadIdx.x >> 5;
    const int lane = threadIdx.x & 31;
    const int hi   = lane >> 4;        // half-wave: K-pair select / C row half
    const int lo   = lane & 15;        // A row / B,C column within tile
    const int kh   = hi * 2;           // K base for this half-wave

    const int row_base = (blockIdx.y * 8 + wave) * 16;   // sample tile
    const int col_base = blockIdx.x * 16;                // output-column tile

    // ---- A operand: x tile, 16x4 f32 (lane lo = row, VGPR pair = K kh,kh+1)
    const float* xr = x + (row_base + lo) * N_IDEAS;
    v2f a0; a0.x = xr[kh];  a0.y = xr[kh + 1];
    // second K-step carries only the 5th input feature in K=0 (lanes 0-15).
    // Load unconditionally (address valid for all lanes), select the value.
    const float x4 = xr[4];
    v2f a1; a1.x = (hi == 0) ? x4 : 0.0f;  a1.y = 0.0f;

    // labels matching the C/D layout: element r of v8f is row (r + 8*hi)
    int lab[8];
#pragma unroll
    for (int r = 0; r < 8; ++r)
        lab[r] = labels[row_base + r + 8 * hi];

    v8f sel = {};  // final selected outputs (every row matches exactly one expert)

#pragma unroll 1
    for (int e = 0; e < N_EXPERTS; ++e) {
        // ---- B operand: W[e]^T tile, 4x16 f32 (lane lo = column, pair = K)
        const float* wr = W + (e * OUT_DIM + (col_base + lo)) * N_IDEAS;
        v2f b0; b0.x = wr[kh];  b0.y = wr[kh + 1];
        const float w4 = wr[4];                       // unconditional load
        v2f b1; b1.x = (hi == 0) ? w4 : 0.0f;  b1.y = 0.0f;  // value select only

        v8f c = {};
        // D = A(16x4) * B(4x16) + C ; K = 0..3
        c = __builtin_amdgcn_wmma_f32_16x16x4_f32(
                /*neg_a=*/false, a0, /*neg_b=*/false, b0,
                /*c_mod=*/(short)0, c, /*reuse_a=*/false, /*reuse_b=*/false);
        // K = 4 (zero padded to 4)
        c = __builtin_amdgcn_wmma_f32_16x16x4_f32(
                false, a1, false, b1, (short)0, c, false, false);

        const float bv = bias[e * OUT_DIM + col_base + lo];
#pragma unroll
        for (int r = 0; r < 8; ++r)
            sel[r] = (lab[r] == e) ? (c[r] + bv) : sel[r];
    }

    // ---- dense coalesced stores: each half-wave writes a contiguous 64B run
#pragma unroll
    for (int r = 0; r < 8; ++r)
        out[(row_base + r + 8 * hi) * OUT_DIM + col_base + lo] = sel[r];
}

extern "C" void kernel_launch(void* const* d_in, const int* in_sizes, int n_in,
                              void* d_out, int out_size, void* d_ws, size_t ws_size,
                              hipStream_t stream) {
    const float* x      = (const float*)d_in[0];
    const int*   labels = (const int*)  d_in[1];
    const float* W      = (const float*)d_in[2];
    const float* bias   = (const float*)d_in[3];
    float* out = (float*)d_out;

    // 49 column tiles; 32768/16 = 2048 row tiles; 8 waves (8 row tiles) per block
    dim3 grid(OUT_DIM / 16, (BATCH / 16) / 8, 1);
    moe_wmma_kernel<<<grid, 256, 0, stream>>>(x, labels, W, bias, out);
}